// MultiModalTreeVQ_42305427865773
// MI455X (gfx1250) — compile-verified
//
#include <hip/hip_runtime.h>
#include <hip/hip_bf16.h>

typedef __attribute__((ext_vector_type(2))) float v2f;
typedef __attribute__((ext_vector_type(4))) float v4f;
typedef __attribute__((ext_vector_type(8))) float v8f;

#define BATCH   8192
#define DIM     300
#define EPAD    304      // padded embedding row stride (floats), 16B-aligned
#define NMOD    6
#define DEPTH   6
#define NROWS   126      // total embedding rows over all levels
#define NRPAD   128
#define ROWS_PER_BLK 128
#define NBLK    ((NMOD*BATCH)/ROWS_PER_BLK)   // 384
#define BLK_PER_M (BATCH/ROWS_PER_BLK)        // 64

__device__ __forceinline__ unsigned order_encode(float f) {
  unsigned u = __float_as_uint(f);
  return (u & 0x80000000u) ? ~u : (u ^ 0x80000000u);
}
__device__ __forceinline__ float order_decode(unsigned u) {
  unsigned b = (u & 0x80000000u) ? (u ^ 0x80000000u) : ~u;
  return __uint_as_float(b);
}

// ---------------------------------------------------------------------------
// Kernel 1: normalize embedding rows into padded table E[128][304]; init aux.
// ---------------------------------------------------------------------------
__global__ void __launch_bounds__(128) prep_kernel(
    const float* __restrict__ e0, const float* __restrict__ e1,
    const float* __restrict__ e2, const float* __restrict__ e3,
    const float* __restrict__ e4, const float* __restrict__ e5,
    float* __restrict__ E, unsigned long long* __restrict__ keysG,
    float* __restrict__ ecPart)
{
  __shared__ float red[128];
  const float* embs[NMOD] = {e0, e1, e2, e3, e4, e5};
  const int r = blockIdx.x;
  const int t = threadIdx.x;

  if (r == 0) {  // init aux buffers (ws is poisoned by harness)
    for (int i = t; i < NMOD * NRPAD; i += blockDim.x) keysG[i] = ~0ull;
    for (int i = t; i < NBLK * DEPTH; i += blockDim.x) ecPart[i] = 0.0f;
  }
  if (r >= NROWS) {
    for (int d = t; d < EPAD; d += blockDim.x) E[(size_t)r * EPAD + d] = 0.0f;
    return;
  }
  const int LOFF[DEPTH] = {0, 2, 6, 14, 30, 62};
  int lvl = 0;
  while (lvl < DEPTH - 1 && r >= LOFF[lvl + 1]) lvl++;
  const float* w = embs[lvl] + (size_t)(r - LOFF[lvl]) * DIM;

  float ss = 0.0f;
  for (int d = t; d < DIM; d += blockDim.x) { float v = w[d]; ss += v * v; }
  red[t] = ss;
  __syncthreads();
  for (int s = 64; s > 0; s >>= 1) {
    if (t < s) red[t] += red[t + s];
    __syncthreads();
  }
  const float inv = 1.0f / fmaxf(sqrtf(red[0]), 1e-12f);
  for (int d = t; d < EPAD; d += blockDim.x)
    E[(size_t)r * EPAD + d] = (d < DIM) ? w[d] * inv : 0.0f;
}

// ---------------------------------------------------------------------------
// Kernel 2: fused f32 WMMA GEMM (S = l2norm(X) @ E^T) + tree routing.
// Block: 256 threads = 8 waves; each wave computes a 16x128 f32 tile with
// V_WMMA_F32_16X16X4_F32; tile kept in LDS, routing done in-place.
// ---------------------------------------------------------------------------
__global__ void __launch_bounds__(256) gemm_route_kernel(
    const float* __restrict__ latents, const float* __restrict__ E,
    unsigned long long* __restrict__ keysG, float* __restrict__ ecPart,
    int* __restrict__ idxAbs, float* __restrict__ s0buf,
    float* __restrict__ outTree)
{
  __shared__ float S[ROWS_PER_BLK * NRPAD];        // 64 KB similarity tile
  __shared__ float invL[ROWS_PER_BLK];
  __shared__ unsigned long long keysL[NRPAD];
  __shared__ float red[256];

  const int t    = threadIdx.x;
  const int wave = t >> 5;        // wave32
  const int lane = t & 31;
  const int hl   = lane >> 4;     // half-wave select (K pair / M-half)
  const int ll   = lane & 15;
  const int rowBase = blockIdx.x * ROWS_PER_BLK;

  // Phase 0: per-row inverse norms (l2norm of x, eps clamp as in reference),
  // b64 vector loads (row stride 1200 B is 8B-aligned).
  if (t < ROWS_PER_BLK) {
    const v2f* xr2 = (const v2f*)(latents + (size_t)(rowBase + t) * DIM);
    float ss = 0.0f;
    for (int d = 0; d < DIM / 2; ++d) {
      v2f v = xr2[d];
      ss += v.x * v.x + v.y * v.y;
    }
    invL[t] = 1.0f / fmaxf(sqrtf(ss), 1e-12f);
  }
  if (t < NROWS) keysL[t] = ~0ull;
  __syncthreads();

  // Phase 1: WMMA main loop. A fragment: lane ll = row M, hl selects K pair.
  const float* aptr = latents + (size_t)(rowBase + wave * 16 + ll) * DIM + 2 * hl;
  v8f zero = {};
  v8f acc[8];
#pragma unroll
  for (int nt = 0; nt < 8; ++nt) acc[nt] = zero;

  for (int k0 = 0; k0 < DIM; k0 += 4) {           // 75 K-steps of 4
    v2f a = *(const v2f*)(aptr + k0);             // 8B aligned
    const float* ebase = E + (size_t)ll * EPAD + k0 + 2 * hl;
#pragma unroll
    for (int nt = 0; nt < 8; ++nt) {              // 8 N-tiles -> 128 cols
      v2f b = *(const v2f*)(ebase + (size_t)nt * 16 * EPAD);
      acc[nt] = __builtin_amdgcn_wmma_f32_16x16x4_f32(
          false, a, false, b, (short)0, acc[nt], false, false);
    }
  }

  // Scale accumulators by 1/||x|| and spill the tile to LDS.
  // C/D layout: VGPR j -> M = j + 8*hl, N = ll (per ISA 16x16 f32 table).
  float invs[8];
#pragma unroll
  for (int j = 0; j < 8; ++j) invs[j] = invL[wave * 16 + 8 * hl + j];
#pragma unroll
  for (int nt = 0; nt < 8; ++nt)
#pragma unroll
    for (int j = 0; j < 8; ++j)
      S[(wave * 16 + 8 * hl + j) * NRPAD + nt * 16 + ll] = acc[nt][j] * invs[j];
  __syncthreads();

  // Phase 2: per-row tree routing (one thread per row).
  const int LOFF[DEPTH] = {0, 2, 6, 14, 30, 62};
  float ec[DEPTH];
#pragma unroll
  for (int i = 0; i < DEPTH; ++i) ec[i] = 0.0f;

  const int gRow = rowBase + t;
  const int m = (rowBase + (t & 127)) >> 13;      // block never straddles modalities
  if (t < ROWS_PER_BLK) {
    const int b = gRow & (BATCH - 1);
    const float* srow = S + t * NRPAD;
    int last = 0;
    for (int i = 0; i < DEPTH; ++i) {
      const int baseRel = (i == 0) ? 0 : 2 * last;
      const int c0 = LOFF[i] + baseRel;
      const float sA = srow[c0], sB = srow[c0 + 1];
      // CE keys: min over (1-s) with first-index tie-break -> pack (-s, b)
      unsigned long long kA = ((unsigned long long)order_encode(-sA) << 32) | (unsigned)b;
      unsigned long long kB = ((unsigned long long)order_encode(-sB) << 32) | (unsigned)b;
      atomicMin(&keysL[c0], kA);
      atomicMin(&keysL[c0 + 1], kB);
      const int pick = (sB > sA) ? 1 : 0;         // argmin distance, tie -> first
      const float sc = pick ? sB : sA;
      const int rel = baseRel + pick;
      ec[i] = sc;
      outTree[(size_t)gRow * DEPTH + i] = (float)rel;
      idxAbs[(size_t)gRow * DEPTH + i] = LOFF[i] + rel;
      last = rel;
    }
    if (b == 0) {  // stash s[b=0, :] for the "no valid row" CE fallback
      for (int c = 0; c < NROWS; ++c) s0buf[m * NRPAD + c] = srow[c];
    }
  }
  __syncthreads();

  // Deterministic EC partial sums: fixed-order LDS tree reduction per level.
  for (int i = 0; i < DEPTH; ++i) {
    red[t] = ec[i];
    __syncthreads();
    for (int s2 = 128; s2 > 0; s2 >>= 1) {
      if (t < s2) red[t] += red[t + s2];
      __syncthreads();
    }
    if (t == 0) ecPart[blockIdx.x * DEPTH + i] = red[0];
    __syncthreads();
  }

  // Push CE keys: one global integer atomicMin per column per block.
  if (t < NROWS) atomicMin(&keysG[(size_t)m * NRPAD + t], keysL[t]);
}

// ---------------------------------------------------------------------------
// Kernel 3: vec gather — the bandwidth-dominant 354 MB write path.
// Vectorized to b128: 300 floats = 75 float4 per (row,level); both the output
// row stride (1200 B) and the E row stride (1216 B) are 16B-aligned.
// ---------------------------------------------------------------------------
__global__ void __launch_bounds__(256) gather_vec_kernel(
    const int* __restrict__ idxAbs, const v4f* __restrict__ E4,
    v4f* __restrict__ outVec4)
{
  const size_t total4 = (size_t)NMOD * BATCH * DEPTH * (DIM / 4);  // 22,118,400
  for (size_t f = (size_t)blockIdx.x * blockDim.x + threadIdx.x; f < total4;
       f += (size_t)gridDim.x * blockDim.x) {
    const int d4 = (int)(f % (DIM / 4));
    const size_t rl = f / (DIM / 4);              // (m*8192+b)*6 + lvl
    const int col = idxAbs[rl];
    outVec4[f] = E4[(size_t)col * (EPAD / 4) + d4];
  }
}

// ---------------------------------------------------------------------------
// Kernel 4: fold partials into the scalar loss (fixed order, deterministic).
// ---------------------------------------------------------------------------
__global__ void loss_kernel(const unsigned long long* __restrict__ keysG,
                            const float* __restrict__ ecPart,
                            const float* __restrict__ s0buf,
                            float* __restrict__ outLoss)
{
  if (threadIdx.x != 0 || blockIdx.x != 0) return;
  const int LOFF[DEPTH] = {0, 2, 6, 14, 30, 62};
  const int LK[DEPTH]   = {2, 4, 8, 16, 32, 64};
  float total = 0.0f;
  for (int m = 0; m < NMOD; ++m) {
    for (int i = 0; i < DEPTH; ++i) {
      float ecsum = 0.0f;
      for (int blk = 0; blk < BLK_PER_M; ++blk)
        ecsum += ecPart[(m * BLK_PER_M + blk) * DEPTH + i];
      const float ecMean = ecsum / (float)BATCH;
      float cesum = 0.0f;
      for (int k = 0; k < LK[i]; ++k) {
        const int c = LOFF[i] + k;
        const unsigned long long key = keysG[(size_t)m * NRPAD + c];
        float val;
        if (key == ~0ull) val = s0buf[m * NRPAD + c];   // all-inf column -> b=0
        else val = -order_decode((unsigned)(key >> 32));
        cesum += val;
      }
      const float ceMean = cesum / (float)LK[i];
      total += 2.0f * (1.0f - ecMean) + 2.0f * (1.0f - ceMean);
    }
  }
  outLoss[0] = total / (float)(NMOD * DEPTH);
}

// ---------------------------------------------------------------------------
extern "C" void kernel_launch(void* const* d_in, const int* in_sizes, int n_in,
                              void* d_out, int out_size, void* d_ws, size_t ws_size,
                              hipStream_t stream)
{
  const float* latents = (const float*)d_in[0];
  const float* e0 = (const float*)d_in[1];
  const float* e1 = (const float*)d_in[2];
  const float* e2 = (const float*)d_in[3];
  const float* e3 = (const float*)d_in[4];
  const float* e4 = (const float*)d_in[5];
  const float* e5 = (const float*)d_in[6];

  float* out = (float*)d_out;
  float* outTree = out;                                           // 6*8192*6
  float* outVec  = out + (size_t)NMOD * BATCH * DEPTH;            // 6*8192*6*300
  float* outLoss = outVec + (size_t)NMOD * BATCH * DEPTH * DIM;   // 1

  char* ws = (char*)d_ws;
  float* E = (float*)ws;                       ws += (size_t)NRPAD * EPAD * sizeof(float);
  unsigned long long* keysG = (unsigned long long*)ws; ws += (size_t)NMOD * NRPAD * sizeof(unsigned long long);
  float* ecPart = (float*)ws;                  ws += (size_t)NBLK * DEPTH * sizeof(float);
  int* idxAbs = (int*)ws;                      ws += (size_t)NMOD * BATCH * DEPTH * sizeof(int);
  float* s0buf = (float*)ws;                   ws += (size_t)NMOD * NRPAD * sizeof(float);

  prep_kernel<<<NRPAD, 128, 0, stream>>>(e0, e1, e2, e3, e4, e5, E, keysG, ecPart);
  gemm_route_kernel<<<NBLK, 256, 0, stream>>>(latents, E, keysG, ecPart,
                                              idxAbs, s0buf, outTree);
  gather_vec_kernel<<<21600, 256, 0, stream>>>(idxAbs, (const v4f*)E, (v4f*)outVec);
  loss_kernel<<<1, 32, 0, stream>>>(keysG, ecPart, s0buf, outLoss);
}